// LapPyramidConv_71296457114370
// MI455X (gfx1250) — compile-verified
//
#include <hip/hip_runtime.h>
#include <cstdint>
#include <cstddef>

// ---------------------------------------------------------------------------
// Laplacian pyramid on MI455X (gfx1250). Bandwidth-bound (~0.007 flop/byte):
//  - async global->LDS staging (CDNA5 GLOBAL_LOAD_ASYNC_TO_LDS_B32/B128 +
//    s_wait_asynccnt) for stencil halo tiles (handles reflect padding since
//    every lane computes its own global address),
//  - float4 (b128) loads/stores on the dominant cur/out streams,
//  - fused blur+downsample and polyphase upsample+subtract,
//  - zero workspace: down-levels staged in the pyr1/pyr2/pyr3 slots of d_out,
//    then overwritten in place (high_k reads cur only at its own element).
// ---------------------------------------------------------------------------

#define NPLANES 24  // B=8 * C=3 depthwise planes

__device__ __forceinline__ int refl(int i, int n) {
    // jax 'reflect' (mirror, edge not repeated), pad <= 2
    if (i < 0) i = -i;
    if (i >= n) i = 2 * n - 2 - i;
    return i;
}

// CDNA5 async copies: global -> LDS, per-lane addresses. Tracked by ASYNCcnt.
__device__ __forceinline__ void async_ld_b32(uint32_t lds_byte_off, const float* g) {
    asm volatile("global_load_async_to_lds_b32 %0, %1, off"
                 :: "v"(lds_byte_off), "v"((unsigned long long)(uintptr_t)g)
                 : "memory");
}
__device__ __forceinline__ void async_ld_b128(uint32_t lds_byte_off, const float* g) {
    asm volatile("global_load_async_to_lds_b128 %0, %1, off"
                 :: "v"(lds_byte_off), "v"((unsigned long long)(uintptr_t)g)
                 : "memory");
}
__device__ __forceinline__ void wait_asynccnt0() {
    asm volatile("s_wait_asynccnt 0x0" ::: "memory");
}
// LDS byte offset = low 32 bits of generic shared address (aperture mapping).
__device__ __forceinline__ uint32_t lds_off(const void* p) {
    return (uint32_t)(uintptr_t)p;
}

// ---------------------------------------------------------------------------
// Kernel A: down[oy][ox] = (gauss5x5 * cur)[2oy][2ox], reflect-padded.
// 16x16 outputs per block -> 36x36 input tile staged in LDS.
// Interior-in-x blocks stage via aligned b128 async loads (40-float rows
// starting at 32*bx-4); edge blocks use per-element b32 with column reflect.
// ---------------------------------------------------------------------------
template <int H>
__global__ __launch_bounds__(256) void blur_down_k(const float* __restrict__ src,
                                                   float* __restrict__ dst) {
    constexpr int W = H, H2 = H / 2, W2 = W / 2;
    __shared__ float tile[36][40];  // col c holds global col (32*bx - 4 + c)
    const int z = blockIdx.z;
    const float* sp = src + (size_t)z * H * W;
    float*       dp = dst + (size_t)z * H2 * W2;

    const int tx = threadIdx.x, ty = threadIdx.y;
    const int tid = ty * 16 + tx;
    const int bx = blockIdx.x, by = blockIdx.y;
    const int base_iy = 32 * by - 2;          // first (unreflected) input row
    const int a0 = 32 * bx - 4;               // aligned first column of tile

    if (bx >= 1 && bx <= (W / 32) - 2) {
        // fully interior in x: 36 rows x 10 x b128
        for (int i = tid; i < 36 * 10; i += 256) {
            const int r = i / 10, c16 = i - r * 10;
            const int gy = refl(base_iy + r, H);
            async_ld_b128(lds_off(&tile[r][c16 * 4]),
                          sp + (size_t)gy * W + (a0 + c16 * 4));
        }
    } else {
        // x-boundary block: scalar loads with column reflect (cols 2..37)
        for (int i = tid; i < 36 * 36; i += 256) {
            const int r = i / 36, c = i - r * 36;
            const int gy = refl(base_iy + r, H);
            const int gx = refl(32 * bx - 2 + c, W);
            async_ld_b32(lds_off(&tile[r][c + 2]), sp + (size_t)gy * W + gx);
        }
    }
    wait_asynccnt0();
    __syncthreads();

    const float w[5] = {1.f, 4.f, 6.f, 4.f, 1.f};
    float acc = 0.f;
#pragma unroll
    for (int r = 0; r < 5; ++r) {
        float h = 0.f;
#pragma unroll
        for (int c = 0; c < 5; ++c)
            h += w[c] * tile[2 * ty + r][2 * tx + c + 2];
        acc += w[r] * h;
    }
    dp[(size_t)(16 * by + ty) * W2 + (16 * bx + tx)] = acc * (1.0f / 256.0f);
}

// ---------------------------------------------------------------------------
// Kernel B: out[y][x] = cur[y][x] - upsample(down)[y][x], 4 x-pixels/thread.
// Polyphase upsample: per axis, even phase taps {1,6,1}/8, odd {4,4}/8.
// Block covers a 64x16 output tile -> 10x34 down tile staged via async b32.
// Safe with out == cur (each thread reads its own float4 before writing it).
// ---------------------------------------------------------------------------
template <int H>
__global__ __launch_bounds__(256) void high_k(const float* __restrict__ cur,
                                              const float* __restrict__ down,
                                              float* __restrict__ out) {
    constexpr int W = H, H2 = H / 2, W2 = W / 2;
    __shared__ float dt[10][36];
    const int z = blockIdx.z;
    const float* cp = cur  + (size_t)z * H * W;
    const float* pp = down + (size_t)z * H2 * W2;
    float*       op = out  + (size_t)z * H * W;

    const int tx = threadIdx.x, ty = threadIdx.y;
    const int tid = ty * 16 + tx;
    const int bx = blockIdx.x, by = blockIdx.y;
    const int base_j = 8 * by - 1, base_i = 32 * bx - 1;

    for (int t = tid; t < 10 * 34; t += 256) {
        const int r = t / 34, c = t - r * 34;
        int j = base_j + r; if (j < 0) j = -j; if (j >= H2) j = H2 - 1;  // zero-stuffed reflect
        int i = base_i + c; if (i < 0) i = -i; if (i >= W2) i = W2 - 1;
        async_ld_b32(lds_off(&dt[r][c]), pp + (size_t)j * W2 + i);
    }
    wait_asynccnt0();
    __syncthreads();

    const int y  = 16 * by + ty;
    const int x0 = 64 * bx + 4 * tx;          // 4 consecutive outputs (even start)
    const int t2y = ty >> 1;

    const float we[3] = {1.f, 6.f, 1.f};
    const float wo[3] = {0.f, 4.f, 4.f};
    const float* wy = (ty & 1) ? wo : we;

    float a0 = 0.f, a1 = 0.f, a2 = 0.f, a3 = 0.f;
#pragma unroll
    for (int r = 0; r < 3; ++r) {
        const float* dr = &dt[t2y + r][2 * tx];
        const float v0 = dr[0], v1 = dr[1], v2 = dr[2], v3 = dr[3];
        const float s0 = v0 + 6.f * v1 + v2;          // x0   (even phase)
        const float s1 = 4.f * (v1 + v2);             // x0+1 (odd phase)
        const float s2 = v1 + 6.f * v2 + v3;          // x0+2 (even phase)
        const float s3 = 4.f * (v2 + v3);             // x0+3 (odd phase)
        const float wr = wy[r];
        a0 += wr * s0; a1 += wr * s1; a2 += wr * s2; a3 += wr * s3;
    }

    const size_t idx = (size_t)y * W + x0;
    const float4 c4 = *reinterpret_cast<const float4*>(cp + idx);
    float4 o4;
    o4.x = c4.x - a0 * (1.0f / 64.0f);
    o4.y = c4.y - a1 * (1.0f / 64.0f);
    o4.z = c4.z - a2 * (1.0f / 64.0f);
    o4.w = c4.w - a3 * (1.0f / 64.0f);
    *reinterpret_cast<float4*>(op + idx) = o4;
}

// ---------------------------------------------------------------------------
// Orchestration: downs into pyr1/pyr2/pyr3 slots, then highs level 0..2
// (pyr1/pyr2 overwritten in place after being consumed). All on `stream`.
// ---------------------------------------------------------------------------
extern "C" void kernel_launch(void* const* d_in, const int* in_sizes, int n_in,
                              void* d_out, int out_size, void* d_ws, size_t ws_size,
                              hipStream_t stream) {
    (void)in_sizes; (void)n_in; (void)out_size; (void)d_ws; (void)ws_size;
    const float* img = (const float*)d_in[0];  // (8,3,1024,1024) f32
    float* out = (float*)d_out;

    const size_t OFF1 = (size_t)NPLANES * 1024 * 1024;
    const size_t OFF2 = OFF1 + (size_t)NPLANES * 512 * 512;
    const size_t OFF3 = OFF2 + (size_t)NPLANES * 256 * 256;

    const dim3 blk(16, 16);

    // Gaussian-blur + downsample chain (down_L stored in pyr_{L+1} slot)
    blur_down_k<1024><<<dim3(1024 / 32, 1024 / 32, NPLANES), blk, 0, stream>>>(img,        out + OFF1);
    blur_down_k< 512><<<dim3( 512 / 32,  512 / 32, NPLANES), blk, 0, stream>>>(out + OFF1, out + OFF2);
    blur_down_k< 256><<<dim3( 256 / 32,  256 / 32, NPLANES), blk, 0, stream>>>(out + OFF2, out + OFF3);

    // High-pass levels: pyr_L = cur_L - upsample(down_L)
    high_k<1024><<<dim3(1024 / 64, 1024 / 16, NPLANES), blk, 0, stream>>>(img,        out + OFF1, out);
    high_k< 512><<<dim3( 512 / 64,  512 / 16, NPLANES), blk, 0, stream>>>(out + OFF1, out + OFF2, out + OFF1);
    high_k< 256><<<dim3( 256 / 64,  256 / 16, NPLANES), blk, 0, stream>>>(out + OFF2, out + OFF3, out + OFF2);
}